// ChebychevTransform_45337674776578
// MI455X (gfx1250) — compile-verified
//
#include <hip/hip_runtime.h>
#include <hip/hip_bf16.h>

typedef __attribute__((ext_vector_type(2))) float v2f;
typedef __attribute__((ext_vector_type(8))) float v8f;

#define B_ 8
#define H_ 256
#define W_ 256
#define C_ 8
#define TILE_W 128
#define LDSW (TILE_W + 3)   // halo: cols w0-1 .. w0+128+1

// Chebyshev (DCT-I style) matrix for n=4, exact rational values:
// T[k][j] = c_k * w_j / 6 * cos(pi*k*j/3), c=[1,2,2,2], w=[1,2,2,1]
__device__ static const float T4[4][4] = {
    {1.f/6.f,  1.f/3.f,  1.f/3.f,  1.f/6.f},
    {1.f/3.f,  1.f/3.f, -1.f/3.f, -1.f/3.f},
    {1.f/3.f, -1.f/3.f, -1.f/3.f,  1.f/3.f},
    {1.f/3.f, -2.f/3.f,  2.f/3.f, -1.f/3.f}};

// out[b,h,w, c*16 + p*4 + k] = sum_{m,n} T[p][m]*T[k][n] * x[b, h+n-1, w+m-1, c] (zero pad)
// GEMM view: Out[pixel, j] = Patch[pixel, f] * K[f, j],  f = m*4+n, j = p*4+k,
//            K[f][j] = T[p][m]*T[k][n].  K-chunk kc of v_wmma_f32_16x16x4 <=> m = kc.
__global__ __launch_bounds__(256) void cheb_wmma_kernel(const float* __restrict__ x,
                                                        float* __restrict__ out) {
  __shared__ float tile[C_][4][LDSW];  // channel-major: conflict-free A-fragment loads

  const int tid = threadIdx.x;
  const int bw = blockIdx.x % (W_ / TILE_W);
  const int h  = (blockIdx.x / (W_ / TILE_W)) % H_;
  const int b  =  blockIdx.x / ((W_ / TILE_W) * H_);
  const int w0 = bw * TILE_W;

  // ---- Stage 4 input rows (h-1..h+2) x 131 cols x 8 ch into LDS, zero-padded ----
  for (int i = tid; i < 4 * LDSW; i += 256) {
    const int rr = i / LDSW, cc = i % LDSW;
    const int gr = h + rr - 1, gc = w0 + cc - 1;
    if (gr >= 0 && gr < H_ && gc >= 0 && gc < W_) {
      const float4* src = (const float4*)(x + (((size_t)b * H_ + gr) * W_ + gc) * C_);
      const float4 lo = src[0], hi = src[1];
      tile[0][rr][cc] = lo.x; tile[1][rr][cc] = lo.y;
      tile[2][rr][cc] = lo.z; tile[3][rr][cc] = lo.w;
      tile[4][rr][cc] = hi.x; tile[5][rr][cc] = hi.y;
      tile[6][rr][cc] = hi.z; tile[7][rr][cc] = hi.w;
    } else {
#pragma unroll
      for (int c = 0; c < C_; ++c) tile[c][rr][cc] = 0.f;
    }
  }
  __syncthreads();

  // ---- Per-wave: 16 pixels x 16 coeffs, one channel per WMMA group ----
  const int wave = tid >> 5;
  const int lane = tid & 31;           // wave32
  const int lidx = lane & 15;
  const int sb   = (lane < 16) ? 0 : 2;   // k_local pair {0,1} or {2,3}  (= n offsets)
  const int wl   = (wave << 4) + lidx;    // pixel index within tile (A: M = lane&15)

  // B fragment (4x16 K-matrix chunk, constant): lane holds B[sb][lidx], B[sb+1][lidx]
  const int p = lidx >> 2, kk = lidx & 3;
  v2f bfrag[4];
#pragma unroll
  for (int kc = 0; kc < 4; ++kc) {
    const float tp = T4[p][kc];
    bfrag[kc].x = tp * T4[kk][sb];
    bfrag[kc].y = tp * T4[kk][sb + 1];
  }

  const size_t pixbase = ((size_t)b * H_ + h) * W_ + w0;
  const int rowadd = (lane < 16) ? 0 : 8;

#pragma unroll
  for (int c = 0; c < C_; ++c) {
    v8f acc = {};
#pragma unroll
    for (int kc = 0; kc < 4; ++kc) {
      // A fragment 16x4: lane holds Patch[pixel=wl][f=kc*4+sb], [f=kc*4+sb+1]
      //   f -> (m=kc, n=sb(+1));  Patch value = tile[c][n][pixel + m]
      v2f a;
      a.x = tile[c][sb][wl + kc];
      a.y = tile[c][sb + 1][wl + kc];
      // 8 args: (neg_a, A, neg_b, B, c_mod, C, reuse_a, reuse_b)
      acc = __builtin_amdgcn_wmma_f32_16x16x4_f32(
          false, a, false, bfrag[kc], (short)0, acc, false, false);
    }
    // D 16x16: VGPR r = (pixel r | r+8, coeff = lane&15) -> 64B coalesced bursts.
    // Output is write-once 256MB (> L2): stream with non-temporal stores.
#pragma unroll
    for (int r = 0; r < 8; ++r) {
      const int px = (wave << 4) + r + rowadd;
      __builtin_nontemporal_store(
          acc[r], &out[(pixbase + px) * (size_t)(C_ * 16) + (c << 4) + lidx]);
    }
  }
}

extern "C" void kernel_launch(void* const* d_in, const int* in_sizes, int n_in,
                              void* d_out, int out_size, void* d_ws, size_t ws_size,
                              hipStream_t stream) {
  (void)in_sizes; (void)n_in; (void)out_size; (void)d_ws; (void)ws_size;
  const float* x = (const float*)d_in[0];
  float* out = (float*)d_out;
  const int blocks = B_ * H_ * (W_ / TILE_W);  // 8*256*2 = 4096
  cheb_wmma_kernel<<<dim3(blocks), dim3(256), 0, stream>>>(x, out);
}